// GCN_11871289606264
// MI455X (gfx1250) — compile-verified
//
#include <hip/hip_runtime.h>
#include <hip/hip_bf16.h>

// ---------------------------------------------------------------------------
// 3-layer GCN on MI455X (gfx1250, wave32).
//  - h = x @ W via V_WMMA_F32_16X16X4_F32 (full-precision matrix pipe)
//  - segment-sum via GLOBAL_ATOMIC_ADD_F32 on float4 chunks (bandwidth bound)
//  - self-loop term folded into accumulator init (no atomics needed for it)
// ---------------------------------------------------------------------------

typedef __attribute__((ext_vector_type(2))) float v2f;
typedef __attribute__((ext_vector_type(8))) float v8f;

#define D 64  // in_dim == hidden_dim == out_dim

// ---- degree / norm precompute ----------------------------------------------

__global__ void gcn_fill_ones(float* __restrict__ deg, int n) {
    int i = blockIdx.x * blockDim.x + threadIdx.x;
    if (i < n) deg[i] = 1.0f;  // self-loop contributes 1 to every node's degree
}

__global__ void gcn_deg_accum(const int* __restrict__ dst, float* __restrict__ deg, int e) {
    int i = blockIdx.x * blockDim.x + threadIdx.x;
    if (i < e) atomicAdd(&deg[dst[i]], 1.0f);
}

__global__ void gcn_rsqrt_inplace(float* __restrict__ deg, int n) {
    int i = blockIdx.x * blockDim.x + threadIdx.x;
    if (i < n) deg[i] = rsqrtf(deg[i]);   // deg >= 1 always (self-loops)
}

__global__ void gcn_edge_norm(const int* __restrict__ src, const int* __restrict__ dst,
                              const float* __restrict__ dinv, float* __restrict__ norm, int e) {
    int i = blockIdx.x * blockDim.x + threadIdx.x;
    if (i < e) norm[i] = dinv[src[i]] * dinv[dst[i]];
}

// ---- dense transform: H[N,64] = X[N,64] @ W[64,64] via WMMA f32 ------------
// Block = 128 threads = 4 waves; one block per 16-row tile; wave w owns the
// 16-column tile [16w, 16w+16). K-loop: 16 steps of V_WMMA_F32_16X16X4_F32.
//
// A 16x4 f32 layout: lane L<16 -> M=L, VGPR{0,1} = K{k0,k0+1};
//                    lane L>=16 -> M=L-16, VGPR{0,1} = K{k0+2,k0+3}.
// B 4x16 f32 layout: lane L<16 -> N=L, K{k0,k0+1}; L>=16 -> N=L-16, K{k0+2,k0+3}.
// C/D 16x16 f32:     lane L<16 -> N=L, VGPRj = M=j; L>=16 -> N=L-16, VGPRj = M=8+j.

__global__ void gcn_gemm_wmma(const float* __restrict__ X, const float* __restrict__ W,
                              float* __restrict__ H) {
    const int rt   = blockIdx.x;            // row tile (16 rows)
    const int wave = threadIdx.x >> 5;      // 0..3 -> column tile
    const int lane = threadIdx.x & 31;
    const int m    = lane & 15;
    const int half = lane >> 4;             // 0: K{+0,+1} / M 0..7 ; 1: K{+2,+3} / M 8..15
    const size_t rowA   = (size_t)rt * 16 + m;
    const int    colB   = wave * 16 + m;

    v8f c = {0.f, 0.f, 0.f, 0.f, 0.f, 0.f, 0.f, 0.f};

    #pragma unroll
    for (int k0 = 0; k0 < D; k0 += 4) {
        const int ka = k0 + half * 2;
        v2f a, b;
        a.x = X[rowA * D + ka];
        a.y = X[rowA * D + ka + 1];
        b.x = W[(size_t)ka       * D + colB];
        b.y = W[(size_t)(ka + 1) * D + colB];
        c = __builtin_amdgcn_wmma_f32_16x16x4_f32(
                /*neg_a=*/false, a, /*neg_b=*/false, b,
                /*c_mod=*/(short)0, c, /*reuse_a=*/false, /*reuse_b=*/false);
    }

    const size_t outRow0 = (size_t)rt * 16 + half * 8;
    #pragma unroll
    for (int j = 0; j < 8; ++j)
        H[(outRow0 + j) * D + colB] = c[j];
}

// Scalar tail for N % 16 rows (not taken for N=100000, kept for generality).
__global__ void gcn_gemm_tail(const float* __restrict__ X, const float* __restrict__ W,
                              float* __restrict__ H, int startRow, int nRows) {
    int idx = blockIdx.x * blockDim.x + threadIdx.x;
    if (idx >= nRows * D) return;
    int r = startRow + idx / D, cc = idx % D;
    float s = 0.f;
    for (int k = 0; k < D; ++k) s += X[(size_t)r * D + k] * W[(size_t)k * D + cc];
    H[(size_t)r * D + cc] = s;
}

// acc[n][d] = bias[d] + dinv[n]^2 * h[n][d]   (bias + self-loop message)
__global__ void gcn_init_acc(const float* __restrict__ h, const float* __restrict__ bias,
                             const float* __restrict__ dinv, float* __restrict__ acc, int n) {
    int idx = blockIdx.x * blockDim.x + threadIdx.x;
    if (idx >= n * D) return;
    int node = idx / D, d = idx % D;
    float di = dinv[node];
    acc[idx] = bias[d] + di * di * h[idx];
}

// Edge scatter: each thread owns one float4 chunk of one edge's message.
__global__ void gcn_edge_scatter(const float* __restrict__ h, const int* __restrict__ src,
                                 const int* __restrict__ dst, const float* __restrict__ norm,
                                 float* __restrict__ acc, long long total) {
    long long gid = (long long)blockIdx.x * blockDim.x + threadIdx.x;
    if (gid >= total) return;
    int e = (int)(gid >> 4);
    int c = (int)(gid & 15) * 4;
    int s = src[e], d = dst[e];
    float nm = norm[e];
    const float4 v = *reinterpret_cast<const float4*>(h + (size_t)s * D + c);
    float* p = acc + (size_t)d * D + c;
    atomicAdd(p + 0, v.x * nm);
    atomicAdd(p + 1, v.y * nm);
    atomicAdd(p + 2, v.z * nm);
    atomicAdd(p + 3, v.w * nm);
}

__global__ void gcn_relu_inplace(float* __restrict__ a, int total) {
    int i = blockIdx.x * blockDim.x + threadIdx.x;
    if (i < total) a[i] = fmaxf(a[i], 0.0f);
}

// ---------------------------------------------------------------------------

extern "C" void kernel_launch(void* const* d_in, const int* in_sizes, int n_in,
                              void* d_out, int out_size, void* d_ws, size_t ws_size,
                              hipStream_t stream) {
    const float* x  = (const float*)d_in[0];
    const int*   ei = (const int*)d_in[1];
    const float* Wm[3] = {(const float*)d_in[2], (const float*)d_in[4], (const float*)d_in[6]};
    const float* bv[3] = {(const float*)d_in[3], (const float*)d_in[5], (const float*)d_in[7]};
    float* out = (float*)d_out;

    const int N = in_sizes[0] / D;
    const int E = in_sizes[1] / 2;
    const int* src = ei;        // edge_index[0]
    const int* dst = ei + E;    // edge_index[1]

    // Workspace carve-up (256B aligned): h | ping | dinv | norm
    char* ws = (char*)d_ws;
    size_t nodeBytes = ((size_t)N * D * sizeof(float) + 255) & ~(size_t)255;
    float* h    = (float*)(ws);
    float* ping = (float*)(ws + nodeBytes);
    float* dinv = (float*)(ws + 2 * nodeBytes);
    size_t dinvBytes = ((size_t)N * sizeof(float) + 255) & ~(size_t)255;
    float* norm = (float*)(ws + 2 * nodeBytes + dinvBytes);

    const int T = 256;
    auto cdiv = [](long long a, long long b) { return (unsigned)((a + b - 1) / b); };

    // --- normalization precompute (once) ---
    gcn_fill_ones<<<cdiv(N, T), T, 0, stream>>>(dinv, N);
    gcn_deg_accum<<<cdiv(E, T), T, 0, stream>>>(dst, dinv, E);
    gcn_rsqrt_inplace<<<cdiv(N, T), T, 0, stream>>>(dinv, N);
    gcn_edge_norm<<<cdiv(E, T), T, 0, stream>>>(src, dst, dinv, norm, E);

    const int tiles = N / 16;
    const int rem   = N - tiles * 16;
    const long long scatterTotal = (long long)E * 16;

    const float* layerIn = x;
    for (int l = 0; l < 3; ++l) {
        float* acc = (l == 2) ? out : ping;
        // dense transform via WMMA (4 waves/block, one 16x16 tile each)
        if (tiles > 0)
            gcn_gemm_wmma<<<tiles, 128, 0, stream>>>(layerIn, Wm[l], h);
        if (rem > 0)
            gcn_gemm_tail<<<cdiv((long long)rem * D, T), T, 0, stream>>>(layerIn, Wm[l], h,
                                                                         tiles * 16, rem);
        // acc = bias + self-loop term
        gcn_init_acc<<<cdiv((long long)N * D, T), T, 0, stream>>>(h, bv[l], dinv, acc, N);
        // edge messages: gather + atomic scatter-add
        gcn_edge_scatter<<<cdiv(scatterTotal, T), T, 0, stream>>>(h, src, dst, norm, acc,
                                                                  scatterTotal);
        // activation
        gcn_relu_inplace<<<cdiv((long long)N * D, T), T, 0, stream>>>(acc, N * D);
        layerIn = acc;
    }
}